// PSFNet_44100724196044
// MI455X (gfx1250) — compile-verified
//
#include <hip/hip_runtime.h>
#include <hip/hip_bf16.h>
#include <math.h>

// Problem constants (match reference)
#define Bz    4
#define Sn    4096
#define Dn    128
#define Kn    13          // n_links
#define Mn    12          // n_W mixing steps
#define Rn    (Bz * Sn)   // 16384 rows total
#define WPAD  16          // K padded to 16 for aligned stores
#define HSTR  132         // LDS row stride (floats); 132 % 64 == 4 -> conflict-free column reads

typedef float v2f __attribute__((ext_vector_type(2)));
typedef float v8f __attribute__((ext_vector_type(8)));

__device__ __forceinline__ float gelu_exact(float v) {
    return 0.5f * v * (1.0f + erff(v * 0.70710678118654752f));
}

__device__ __forceinline__ v8f wmma_f32x4(v2f a, v2f b, v8f c) {
    // V_WMMA_F32_16X16X4_F32 : D = A(16x4 f32) * B(4x16 f32) + C(16x16 f32)
    return __builtin_amdgcn_wmma_f32_16x16x4_f32(
        /*neg_a=*/false, a, /*neg_b=*/false, b,
        /*c_mod=*/(short)0, c, /*reuse_a=*/false, /*reuse_b=*/false);
}

// ---------------------------------------------------------------------------
// x[b,s,:] = emb[data[b,s],:] + apc[s,:]     (float4 vectorized)
// ---------------------------------------------------------------------------
__global__ __launch_bounds__(256) void embed_kernel(
    const int* __restrict__ data, const float* __restrict__ emb,
    const float* __restrict__ apc, float* __restrict__ x)
{
    int tid = blockIdx.x * blockDim.x + threadIdx.x;      // one thread = 4 floats
    int row = tid >> 5;                                   // Dn/4 == 32 chunks per row
    int c4  = tid & 31;
    if (row >= Rn) return;
    int s   = row & (Sn - 1);
    int tok = data[row];
    const float4* e = (const float4*)(emb + (size_t)tok * Dn);
    const float4* a = (const float4*)(apc + (size_t)s * Dn);
    float4 ev = e[c4], av = a[c4];
    float4 o; o.x = ev.x + av.x; o.y = ev.y + av.y; o.z = ev.z + av.z; o.w = ev.w + av.w;
    ((float4*)x)[tid] = o;
}

// ---------------------------------------------------------------------------
// V = gelu(x @ W1 + b1) @ W2 + b2   (one wave -> 16 rows x 128 cols)
// ---------------------------------------------------------------------------
__global__ __launch_bounds__(128) void gmlp_kernel(
    const float* __restrict__ x,
    const float* __restrict__ W1, const float* __restrict__ b1,
    const float* __restrict__ W2, const float* __restrict__ b2,
    float* __restrict__ outV)
{
    __shared__ float Hlds[4][16 * HSTR];
    const int wave    = threadIdx.x >> 5;
    const int lane    = threadIdx.x & 31;
    const int half    = lane >> 4;        // 0: K/row 0,1 ; 1: K/row 2,3
    const int mn      = lane & 15;        // M for A-frags, N for B/C/D-frags
    const int rowBase = blockIdx.x * 64 + wave * 16;

    // ---- layer 1: H = gelu(X @ W1 + b1) ----
    v8f acc[8];
    #pragma unroll
    for (int t = 0; t < 8; t++) acc[t] = {};
    const float* Xrow = x + (size_t)(rowBase + mn) * Dn;
    #pragma unroll 4
    for (int k = 0; k < Dn; k += 4) {
        float2 av = *(const float2*)(Xrow + k + 2 * half);
        v2f a; a.x = av.x; a.y = av.y;
        #pragma unroll
        for (int t = 0; t < 8; t++) {
            const float* wp = W1 + (size_t)(k + 2 * half) * Dn + t * 16 + mn;
            v2f b; b.x = wp[0]; b.y = wp[Dn];
            acc[t] = wmma_f32x4(a, b, acc[t]);
        }
    }
    #pragma unroll
    for (int t = 0; t < 8; t++) {
        float bias = b1[t * 16 + mn];
        #pragma unroll
        for (int r = 0; r < 8; r++) {
            Hlds[wave][(r + 8 * half) * HSTR + t * 16 + mn] =
                gelu_exact(acc[t][r] + bias);
        }
    }
    __syncthreads();

    // ---- layer 2: V = H @ W2 + b2 ----
    v8f acc2[8];
    #pragma unroll
    for (int t = 0; t < 8; t++) acc2[t] = {};
    #pragma unroll 4
    for (int k = 0; k < Dn; k += 4) {
        float2 av = *(const float2*)(&Hlds[wave][mn * HSTR + k + 2 * half]);
        v2f a; a.x = av.x; a.y = av.y;
        #pragma unroll
        for (int t = 0; t < 8; t++) {
            const float* wp = W2 + (size_t)(k + 2 * half) * Dn + t * 16 + mn;
            v2f b; b.x = wp[0]; b.y = wp[Dn];
            acc2[t] = wmma_f32x4(a, b, acc2[t]);
        }
    }
    #pragma unroll
    for (int t = 0; t < 8; t++) {
        float bias = b2[t * 16 + mn];
        #pragma unroll
        for (int r = 0; r < 8; r++) {
            outV[(size_t)(rowBase + r + 8 * half) * Dn + t * 16 + mn] =
                acc2[t][r] + bias;
        }
    }
}

// ---------------------------------------------------------------------------
// W_all[m] = gelu(x @ fW1[m] + fb1[m]) @ fW2[m] + fb2[m]   (N padded 13->16)
// ---------------------------------------------------------------------------
__global__ __launch_bounds__(128) void fmlp_kernel(
    const float* __restrict__ x,
    const float* __restrict__ fW1, const float* __restrict__ fb1,
    const float* __restrict__ fW2, const float* __restrict__ fb2,
    float* __restrict__ Wpad)
{
    __shared__ float Hlds[4][16 * HSTR];
    const int m       = blockIdx.y;
    const int wave    = threadIdx.x >> 5;
    const int lane    = threadIdx.x & 31;
    const int half    = lane >> 4;
    const int mn      = lane & 15;
    const int rowBase = blockIdx.x * 64 + wave * 16;

    const float* W1 = fW1 + (size_t)m * Dn * Dn;
    const float* B1 = fb1 + (size_t)m * Dn;
    const float* W2 = fW2 + (size_t)m * Dn * Kn;
    const float* B2 = fb2 + (size_t)m * Kn;

    // ---- layer 1 ----
    v8f acc[8];
    #pragma unroll
    for (int t = 0; t < 8; t++) acc[t] = {};
    const float* Xrow = x + (size_t)(rowBase + mn) * Dn;
    #pragma unroll 4
    for (int k = 0; k < Dn; k += 4) {
        float2 av = *(const float2*)(Xrow + k + 2 * half);
        v2f a; a.x = av.x; a.y = av.y;
        #pragma unroll
        for (int t = 0; t < 8; t++) {
            const float* wp = W1 + (size_t)(k + 2 * half) * Dn + t * 16 + mn;
            v2f b; b.x = wp[0]; b.y = wp[Dn];
            acc[t] = wmma_f32x4(a, b, acc[t]);
        }
    }
    #pragma unroll
    for (int t = 0; t < 8; t++) {
        float bias = B1[t * 16 + mn];
        #pragma unroll
        for (int r = 0; r < 8; r++) {
            Hlds[wave][(r + 8 * half) * HSTR + t * 16 + mn] =
                gelu_exact(acc[t][r] + bias);
        }
    }
    __syncthreads();

    // ---- layer 2: single 16-wide N tile, cols >= 13 are zero ----
    v8f acc2 = {};
    const bool valid = (mn < Kn);
    #pragma unroll 4
    for (int k = 0; k < Dn; k += 4) {
        float2 av = *(const float2*)(&Hlds[wave][mn * HSTR + k + 2 * half]);
        v2f a; a.x = av.x; a.y = av.y;
        v2f b;
        b.x = valid ? W2[(size_t)(k + 2 * half) * Kn + mn]     : 0.0f;
        b.y = valid ? W2[(size_t)(k + 2 * half + 1) * Kn + mn] : 0.0f;
        acc2 = wmma_f32x4(a, b, acc2);
    }
    float bias = valid ? B2[mn] : 0.0f;
    #pragma unroll
    for (int r = 0; r < 8; r++) {
        Wpad[((size_t)m * Rn + rowBase + r + 8 * half) * WPAD + mn] =
            acc2[r] + bias;
    }
}

// ---------------------------------------------------------------------------
// Chord mixing step: dst[b,i,:] = sum_k w[b,i,k] * src[b,(i+off_k)%S,:] + res
// One wave per row; lanes cover D with float4.
// ---------------------------------------------------------------------------
__global__ __launch_bounds__(256) void chord_kernel(
    const float* __restrict__ Vsrc, const float* __restrict__ res,
    const float* __restrict__ Wm, float* __restrict__ Vdst)
{
    const int row  = blockIdx.x * 8 + (threadIdx.x >> 5);
    const int lane = threadIdx.x & 31;
    const int b    = row >> 12;          // row / Sn
    const int i    = row & (Sn - 1);

    const int offs[Kn] = {0, 1, 2, 4, 8, 16, 32, 64, 128, 256, 512, 1024, 2048};

    float w[Kn];
    const float* wrow = Wm + (size_t)row * WPAD;
    #pragma unroll
    for (int k = 0; k < Kn; k++) w[k] = wrow[k];

    const float4* srcb = (const float4*)(Vsrc + (size_t)b * Sn * Dn);
    float4 acc = ((const float4*)(res + (size_t)row * Dn))[lane];
    #pragma unroll
    for (int k = 0; k < Kn; k++) {
        int j = (i + offs[k]) & (Sn - 1);
        float4 v = srcb[(size_t)j * (Dn / 4) + lane];
        acc.x += w[k] * v.x; acc.y += w[k] * v.y;
        acc.z += w[k] * v.z; acc.w += w[k] * v.w;
    }
    ((float4*)(Vdst + (size_t)row * Dn))[lane] = acc;
}

// ---------------------------------------------------------------------------
extern "C" void kernel_launch(void* const* d_in, const int* in_sizes, int n_in,
                              void* d_out, int out_size, void* d_ws, size_t ws_size,
                              hipStream_t stream) {
    const int*   data = (const int*)  d_in[0];
    /* d_in[1] = cols : recomputed analytically on device */
    const float* emb  = (const float*)d_in[2];
    const float* apc  = (const float*)d_in[3];
    const float* gW1  = (const float*)d_in[4];
    const float* gb1  = (const float*)d_in[5];
    const float* gW2  = (const float*)d_in[6];
    const float* gb2  = (const float*)d_in[7];
    const float* fW1  = (const float*)d_in[8];
    const float* fb1  = (const float*)d_in[9];
    const float* fW2  = (const float*)d_in[10];
    const float* fb2  = (const float*)d_in[11];
    float* out = (float*)d_out;

    // Workspace layout (all f32):
    //   x    : Rn*Dn       =  8 MiB
    //   res  : Rn*Dn       =  8 MiB  (initial V, re-added every step)
    //   Va   : Rn*Dn       =  8 MiB  (ping-pong partner of d_out)
    //   Wpad : Mn*Rn*WPAD  = 12 MiB
    char*  ws   = (char*)d_ws;
    float* xbuf = (float*)(ws);
    float* res  = (float*)(ws + (size_t) 8 * 1024 * 1024);
    float* Va   = (float*)(ws + (size_t)16 * 1024 * 1024);
    float* Wpad = (float*)(ws + (size_t)24 * 1024 * 1024);

    embed_kernel<<<(Rn * (Dn / 4)) / 256, 256, 0, stream>>>(data, emb, apc, xbuf);
    gmlp_kernel <<<Rn / 64, 128, 0, stream>>>(xbuf, gW1, gb1, gW2, gb2, res);
    fmlp_kernel <<<dim3(Rn / 64, Mn), 128, 0, stream>>>(xbuf, fW1, fb1, fW2, fb2, Wpad);

    const float* src = res;
    for (int m = 0; m < Mn; m++) {
        float* dst = (m & 1) ? out : Va;   // m=11 (last) lands in d_out
        chord_kernel<<<Rn / 8, 256, 0, stream>>>(
            src, res, Wpad + (size_t)m * Rn * WPAD, dst);
        src = dst;
    }
}